// AdaptiveFrequencySelector_37469294690510
// MI455X (gfx1250) — compile-verified
//
#include <hip/hip_runtime.h>

typedef __attribute__((ext_vector_type(2))) float v2f;
typedef __attribute__((ext_vector_type(8))) float v8f;

#define B_  8
#define F_  4096
#define D_  1024
#define H_  8
#define WSTRIDE (D_ + 2)   // bank-conflict-free LDS stride for transposed W

// ---------------------------------------------------------------------------
// Pass 1: scores(b,h,f) = sum_d |x[b,f,d]| * W[d,h] + bias[h]
// One wave computes a 16(M=f) x 16(N=h) tile via V_WMMA_F32_16X16X4_F32.
// A layout (ISA 7.12.2, 32-bit A 16x4): lane l -> row M=l&15, K = v + 2*(l>>4).
// B layout (rows striped across lanes):  lane l -> col N=l&15, K = v + 2*(l>>4).
// Columns 8..15 duplicate columns 0..7 (unconditional loads; results discarded)
// so no divergent/zero-padded B loads are needed.
// C layout: lane l -> col N=l&15, VGPR r -> row M = r + 8*(l>>4).
// ---------------------------------------------------------------------------
__global__ __launch_bounds__(256) void scores_wmma_kernel(
    const float* __restrict__ x, const float* __restrict__ W,
    const float* __restrict__ bias, float* __restrict__ scores /* (B,H,F) */) {
  __shared__ float Wt[H_ * WSTRIDE];  // transposed W: Wt[h][d], ~32 KB
  __shared__ float bl[H_];
  for (int i = threadIdx.x; i < D_ * H_; i += 256) {  // coalesced read, LDS scatter
    const int d = i >> 3;
    const int c = i & 7;
    Wt[c * WSTRIDE + d] = W[i];
  }
  if (threadIdx.x < H_) bl[threadIdx.x] = bias[threadIdx.x];
  __syncthreads();

  const int wave = threadIdx.x >> 5;
  const int lane = threadIdx.x & 31;
  const int half = lane >> 4;       // K sub-group
  const int col  = lane & 15;       // A-row index == C-column index
  const int colw = col & 7;         // duplicated W column for lanes 8..15

  const int m0 = (blockIdx.x * 8 + wave) * 16;   // global row base (b*F + f)
  const int b  = m0 >> 12;                       // / F_
  const int f0 = m0 & (F_ - 1);

  const float* xrow = x + ((size_t)b * F_ + (f0 + col)) * D_;
  const float* wrow = &Wt[colw * WSTRIDE];

  v8f acc = {};
  for (int kk = 0; kk < D_; kk += 64) {
    if (kk + 64 < D_) __builtin_prefetch(xrow + kk + 64, 0, 3); // global_prefetch_b8
#pragma unroll
    for (int k0 = kk; k0 < kk + 64; k0 += 4) {
      const int d0 = k0 + 2 * half;
      const float2 xv = *(const float2*)(xrow + d0);   // global_load_b64
      const float2 wv = *(const float2*)(wrow + d0);   // ds_load_b64
      v2f a;
      a.x = fabsf(xv.x);
      a.y = fabsf(xv.y);
      v2f bm;
      bm.x = wv.x;
      bm.y = wv.y;
      acc = __builtin_amdgcn_wmma_f32_16x16x4_f32(false, a, false, bm,
                                                  (short)0, acc, false, false);
    }
  }

  if (col < H_) {
    const float bia = bl[col];
    float* srow = scores + ((size_t)b * H_ + col) * F_;
#pragma unroll
    for (int r = 0; r < 8; ++r)
      srow[f0 + r + 8 * half] = acc[r] + bia;
  }
}

// ---------------------------------------------------------------------------
// Pass 2: per (b,h) exact top-k threshold + mask (ties -> lowest index, like
// jax.lax.top_k). One block per (b,h); MSB-first radix select over uint-order
// transform of the 4096 scores held in LDS.
// ---------------------------------------------------------------------------
__global__ __launch_bounds__(256) void topk_mask_kernel(
    const float* __restrict__ scores, const float* __restrict__ soff,
    float* __restrict__ mask /* (B,F,H) */) {
  const int b = blockIdx.x >> 3;
  const int h = blockIdx.x & 7;

  // per-head k: float64 to match Python's trace-time int() truncation
  const double off = (double)soff[h];
  const double sp  = 1.0 / (1.0 + exp(-off)) * 0.3 + 0.15;
  int k = (int)((double)F_ * sp);
  if (k < 1)  k = 1;
  if (k > F_) k = F_;

  __shared__ unsigned uarr[F_];     // 16 KB
  __shared__ unsigned hist[256];
  __shared__ unsigned sh_prefix;
  __shared__ int      sh_rem;

  const float* srow = scores + ((size_t)b * H_ + h) * F_;
  for (int i = threadIdx.x; i < F_; i += 256) {
    unsigned u = __float_as_uint(srow[i]);
    u = (u & 0x80000000u) ? ~u : (u | 0x80000000u);  // order-preserving
    uarr[i] = u;
  }
  if (threadIdx.x == 0) { sh_prefix = 0u; sh_rem = k; }
  __syncthreads();

  for (int p = 3; p >= 0; --p) {
    hist[threadIdx.x] = 0u;
    __syncthreads();
    const unsigned hi_mask = (p == 3) ? 0u : (0xFFFFFFFFu << (8 * (p + 1)));
    const unsigned pref = sh_prefix;
    for (int i = threadIdx.x; i < F_; i += 256) {
      const unsigned u = uarr[i];
      if ((u & hi_mask) == (pref & hi_mask))
        atomicAdd(&hist[(u >> (8 * p)) & 255u], 1u);
    }
    __syncthreads();
    if (threadIdx.x == 0) {
      int rem = sh_rem;
      unsigned cum = 0;
      for (int bin = 255; bin >= 0; --bin) {
        cum += hist[bin];
        if ((int)cum >= rem) {
          sh_prefix = pref | ((unsigned)bin << (8 * p));
          sh_rem = rem - (int)(cum - hist[bin]);
          break;
        }
      }
    }
    __syncthreads();
  }

  const unsigned T   = sh_prefix;  // k-th largest (uint order)
  const int     r_eq = sh_rem;     // how many elements == T to take (in index order)

  if (threadIdx.x < 32) {          // one wave32, index-ordered tie resolution
    const int lane = threadIdx.x;
    int taken = 0;
    for (int base = 0; base < F_; base += 32) {
      const unsigned u  = uarr[base + lane];
      const bool     eq = (u == T);
      const unsigned long long bal = __ballot(eq);
      const int pre = __popcll(bal & ((1ull << lane) - 1ull));
      float m = (u > T) ? 1.0f : ((eq && (taken + pre) < r_eq) ? 1.0f : 0.0f);
      mask[(((size_t)b * F_ + base + lane) * H_) + h] = m;
      taken += __popcll(bal);
    }
  }
}

// ---------------------------------------------------------------------------
// Pass 3: out = x * mask (head_dim = 128 broadcast), float4-vectorized.
// x should still be largely resident in the 192 MB L2 after pass 1.
// ---------------------------------------------------------------------------
__global__ __launch_bounds__(256) void apply_mask_kernel(
    const float* __restrict__ x, const float* __restrict__ mask,
    float* __restrict__ out) {
  const size_t idx = (size_t)blockIdx.x * 256 + threadIdx.x; // float4 index
  const size_t e   = idx * 4;
  const int d = (int)(e & (D_ - 1));
  const int f = (int)((e >> 10) & (F_ - 1));
  const int b = (int)(e >> 22);
  const int h = d >> 7;                       // head_dim = 128
  const float m = mask[(((size_t)b * F_ + f) * H_) + h];
  const float4 xv = ((const float4*)x)[idx];
  float4 o;
  o.x = xv.x * m; o.y = xv.y * m; o.z = xv.z * m; o.w = xv.w * m;
  ((float4*)out)[idx] = o;
}

extern "C" void kernel_launch(void* const* d_in, const int* in_sizes, int n_in,
                              void* d_out, int out_size, void* d_ws, size_t ws_size,
                              hipStream_t stream) {
  const float* x    = (const float*)d_in[0];   // (B,F,D) f32
  const float* W    = (const float*)d_in[1];   // (D,H)   f32
  const float* bias = (const float*)d_in[2];   // (H,)    f32
  const float* soff = (const float*)d_in[3];   // (H,)    f32

  float* out      = (float*)d_out;                       // x_filtered (B,F,D)
  float* out_mask = out + (size_t)B_ * F_ * D_;          // mask (B,F,H)
  float* scores   = (float*)d_ws;                        // (B,H,F) = 1 MiB scratch

  scores_wmma_kernel<<<(B_ * F_) / 128, 256, 0, stream>>>(x, W, bias, scores);
  topk_mask_kernel<<<B_ * H_, 256, 0, stream>>>(scores, soff, out_mask);
  apply_mask_kernel<<<((size_t)B_ * F_ * D_ / 4) / 256, 256, 0, stream>>>(x, out_mask, out);
}